// SimpleDecoder_86758339379420
// MI455X (gfx1250) — compile-verified
//
#include <hip/hip_runtime.h>

// Problem constants (reference: B=128, G=28, C5=85)
#define BB   128
#define GG   28
#define NBOX 784          // G*G
#define C5   85
#define NPAD 1024         // bitonic-sort padded size
#define OBJ_THR 0.1f
#define NMS_THR 0.5f

#define SLAB_ELE  (NBOX * C5)          // 66640 floats = 266560 B per batch
#define SLAB_BYTE (SLAB_ELE * 4)
#define HALF_BYTE (SLAB_BYTE / 2)      // 133280 B : TDM half / async half
#define HALF_ELE  (SLAB_ELE / 2)       // 33320 elements = 392 rows of 85

// Dynamic-LDS layout. [0, SLAB_BYTE) holds the pred slab during load+decode;
// after decode it is dead and the sort/NMS scratch aliases it.
#define OFF_SKEY  0u                   // float[1024]           (aliases slab)
#define OFF_SIDX  4096u                // int[1024]             (aliases slab)
#define OFF_SX1   8192u                // float[784] x5         (aliases slab)
#define OFF_SY1   11328u
#define OFF_SX2   14464u
#define OFF_SY2   17600u
#define OFF_SAR   20736u
#define OFF_ADJ   23872u               // ushort[16][52]        (aliases slab)
#define OFF_SUPP  25536u               // uint[32]              (aliases slab)
#define OFF_SDEC  ((unsigned)SLAB_BYTE)        // float[784*6], persistent
#define OFF_NV    (OFF_SDEC + (unsigned)(NBOX * 6 * 4))
#define SMEM_BYTES (OFF_NV + 16u)      // 285392 B  (< 320KB WGP LDS)

typedef __attribute__((ext_vector_type(2))) float v2f;
typedef __attribute__((ext_vector_type(8))) float v8f;
typedef __attribute__((ext_vector_type(4))) unsigned int u32x4;
typedef __attribute__((ext_vector_type(8))) unsigned int u32x8;

// ---------------------------------------------------------------------------
// Fused kernel: one 256-thread block per batch.
//  stage 0: TDM streams rows 0..391 of the pred slab to LDS, waves async-DMA
//           rows 392..783 (global_load_async_to_lds_b128)
//  stage 1: wave-per-cell decode (shuffle max/argmax over 80 classes)
//  stage 2: stable bitonic sort of (score desc, idx asc) — argsort(-score)
//  stage 3: gather sorted boxes -> global + SoA in LDS
//  stage 4: 16-row strips; v_wmma_f32_16x16x4_f32 computes the pairwise
//           union tile (area_i + area_j - inter via C = -inter); greedy apply
//  stage 5: keep mask
// ---------------------------------------------------------------------------
__global__ __launch_bounds__(256) void yolo_fused_kernel(
    const float* __restrict__ pred, float* __restrict__ boxesOut,
    float* __restrict__ keepOut) {
  extern __shared__ __attribute__((aligned(16))) unsigned char smem[];

  float*          slab  = (float*)smem;                       // load/decode phase
  float*          skey  = (float*)(smem + OFF_SKEY);          // sort phase
  int*            sidxA = (int*)(smem + OFF_SIDX);
  float*          sx1   = (float*)(smem + OFF_SX1);
  float*          sy1   = (float*)(smem + OFF_SY1);
  float*          sx2   = (float*)(smem + OFF_SX2);
  float*          sy2   = (float*)(smem + OFF_SY2);
  float*          sar   = (float*)(smem + OFF_SAR);
  unsigned short (*adj16)[52] = (unsigned short (*)[52])(smem + OFF_ADJ);
  unsigned int*   suppW = (unsigned int*)(smem + OFF_SUPP);
  float*          sdec  = (float*)(smem + OFF_SDEC);
  int*            nvp   = (int*)(smem + OFF_NV);

  const int tid  = threadIdx.x;
  const int lane = tid & 31;
  const int wv   = tid >> 5;
  const int b    = blockIdx.x;
  const float* predB = pred + (size_t)b * SLAB_ELE;

  if (tid == 0) *nvp = 0;              // OFF_NV is outside the DMA'd region

  // ---- Stage 0: fill the slab. TDM half + async-DMA half ------------------
  const unsigned ldsBase = (unsigned)(uintptr_t)&slab[0];  // addr[31:0] == LDS offset
  if (wv == 0) {
    // Tensor DMA descriptor: 2D tile 4760 x 7 f32 (= 392 rows of 85), dense.
    const unsigned long long ga = (unsigned long long)(uintptr_t)predB;
    u32x4 g0;
    g0.x = 1u;                                         // count=1, user mode
    g0.y = ldsBase;                                    // lds_addr
    g0.z = (unsigned)(ga & 0xFFFFFFFFull);             // global_addr[31:0]
    g0.w = (unsigned)((ga >> 32) & 0x01FFFFFFull)      // global_addr[56:32]
         | (2u << 30);                                 // type = 2 (image)
    u32x8 g1;
    g1.s0 = (2u << 16);                                // data_size = 4B
    g1.s1 = (4760u & 0xFFFFu) << 16;                   // tensor_dim0 lo16
    g1.s2 = (7u & 0xFFFFu) << 16;                      // dim0 hi16=0 | tensor_dim1 lo16
    g1.s3 = (4760u << 16);                             // dim1 hi16=0 | tile_dim0
    g1.s4 = 7u;                                        // tile_dim1 | tile_dim2=0
    g1.s5 = 4760u;                                     // tensor_dim0_stride lo32
    g1.s6 = 0u;                                        // stride hi | dim1_stride lo
    g1.s7 = 0u;
    u32x4 gz = {0u, 0u, 0u, 0u};                       // groups 2/3: unused dims
    asm volatile("tensor_load_to_lds %0, %1, %2, %3"
                 :: "s"(g0), "s"(g1), "s"(gz), "s"(gz) : "memory");
  }
  {
    // Async half: bytes [HALF_BYTE, SLAB_BYTE), 8330 x b128 across 256 threads
    const unsigned ldsA = ldsBase + (unsigned)HALF_BYTE;
    const unsigned long long gaA =
        (unsigned long long)(uintptr_t)(predB + HALF_ELE);
    for (int c = tid; c < HALF_BYTE / 16; c += 256) {
      unsigned l = ldsA + (unsigned)c * 16u;
      unsigned long long ga = gaA + (unsigned long long)c * 16ull;
      asm volatile("global_load_async_to_lds_b128 %0, %1, off"
                   :: "v"(l), "v"(ga) : "memory");
    }
  }
  asm volatile("s_wait_asynccnt 0" ::: "memory");
  if (wv == 0) __builtin_amdgcn_s_wait_tensorcnt(0);
  __syncthreads();

  // ---- Stage 1: decode. Wave wv handles cells wv, wv+8, ... ---------------
  for (int t = 0; t < NBOX / 8; ++t) {
    const int cell = (t << 3) | wv;
    const float* p = slab + cell * C5;
    float v0 = p[lane];
    float v1 = p[lane + 32];
    float v2 = (lane + 64 < C5) ? p[lane + 64] : -1.0f;

    float m  = (lane >= 5) ? v0 : -1.0f;
    int   mi = lane - 5;                 // first-occurrence argmax tie-break
    if (v1 > m) { m = v1; mi = lane + 27; }
    if (v2 > m) { m = v2; mi = lane + 59; }
    #pragma unroll
    for (int off = 16; off > 0; off >>= 1) {
      float om = __shfl_xor(m, off, 32);
      int   oi = __shfl_xor(mi, off, 32);
      if (om > m || (om == m && oi < mi)) { m = om; mi = oi; }
    }
    const float bx = __shfl(v0, 0, 32);
    const float by = __shfl(v0, 1, 32);
    const float bw = __shfl(v0, 2, 32);
    const float bh = __shfl(v0, 3, 32);
    const float ct = __shfl(v0, 4, 32);

    const int gi = cell / GG, gj = cell % GG;   // pred axes 1,2
    const float cx = (bx + (float)gi) / (float)GG;
    const float cy = (by + (float)gj) / (float)GG;
    const float score = ct * m;
    const float msk   = (score > OBJ_THR) ? 1.0f : 0.0f;

    float o = (cx - 0.5f * bw) * msk;                 // lane 0: x1
    if (lane == 1) o = (cy - 0.5f * bh) * msk;        // y1
    if (lane == 2) o = (cx + 0.5f * bw) * msk;        // x2
    if (lane == 3) o = (cy + 0.5f * bh) * msk;        // y2
    if (lane == 4) o = (float)mi * msk;               // class idx
    if (lane == 5) o = score * msk;                   // prob
    if (lane < 6) sdec[cell * 6 + lane] = o;
  }
  __syncthreads();   // slab is dead past this point; scratch may alias it

  // ---- Stage 2a: keys (score desc, idx asc), count valid ------------------
  {
    int cnt = 0;
    for (int i = tid; i < NPAD; i += 256) {
      float s = -__builtin_inff();
      if (i < NBOX) {
        float sc = sdec[i * 6 + 5];
        if (sc > OBJ_THR) { s = sc; cnt++; }
      }
      skey[i]  = s;
      sidxA[i] = i;
    }
    atomicAdd(nvp, cnt);
    __syncthreads();
  }

  // ---- Stage 2b: stable bitonic sort, best-first --------------------------
  for (int k = 2; k <= NPAD; k <<= 1) {
    for (int j = k >> 1; j > 0; j >>= 1) {
      for (int i = tid; i < NPAD; i += 256) {
        int ixj = i ^ j;
        if (ixj > i) {
          float s1 = skey[i], s2 = skey[ixj];
          int   i1 = sidxA[i], i2 = sidxA[ixj];
          bool dir = ((i & k) == 0);
          bool bef = (s2 > s1) || (s2 == s1 && i2 < i1);
          if (bef == dir) {
            skey[i] = s2; skey[ixj] = s1;
            sidxA[i] = i2; sidxA[ixj] = i1;
          }
        }
      }
      __syncthreads();
    }
  }

  // ---- Stage 3: gather sorted boxes -> global out + SoA -------------------
  float* outB = boxesOut + (size_t)b * NBOX * 6;
  for (int jj = tid; jj < NBOX; jj += 256) {
    int src = sidxA[jj];
    float bx1 = sdec[src * 6 + 0], by1 = sdec[src * 6 + 1];
    float bx2 = sdec[src * 6 + 2], by2 = sdec[src * 6 + 3];
    float bcl = sdec[src * 6 + 4], bsc = sdec[src * 6 + 5];
    outB[jj * 6 + 0] = bx1; outB[jj * 6 + 1] = by1;
    outB[jj * 6 + 2] = bx2; outB[jj * 6 + 3] = by2;
    outB[jj * 6 + 4] = bcl; outB[jj * 6 + 5] = bsc;
    sx1[jj] = bx1; sy1[jj] = by1; sx2[jj] = bx2; sy2[jj] = by2;
    sar[jj] = (bx2 - bx1) * (by2 - by1);
  }
  if (tid < 32) suppW[tid] = 0u;
  __syncthreads();

  const int nv    = *nvp;            // valid boxes are positions [0, nv)
  const int jtMax = (nv + 15) >> 4;

  // ---- Stage 4: strips of 16 rows; WMMA union tiles + greedy apply --------
  for (int i0 = 0; i0 < nv; i0 += 16) {
    const int rb = (lane < 16) ? 0 : 8;     // C-layout rows for this lane
    float rx1[8], ry1[8], rx2[8], ry2[8];
    #pragma unroll
    for (int r = 0; r < 8; ++r) {
      int i = i0 + rb + r;                  // i0 <= 768 -> i <= 783
      rx1[r] = sx1[i]; ry1[r] = sy1[i]; rx2[r] = sx2[i]; ry2[r] = sy2[i];
    }
    v2f A;                                  // A(16x4): [area_row, 1, 0, 0]
    A.x = (lane < 16) ? sar[i0 + lane] : 0.0f;
    A.y = (lane < 16) ? 1.0f : 0.0f;

    for (int jt = wv; jt < jtMax; jt += 8) {
      const int jc = jt * 16 + (lane & 15);
      const float cx1 = sx1[jc], cy1 = sy1[jc];
      const float cx2 = sx2[jc], cy2 = sy2[jc];
      v2f Bv;                               // B(4x16): [1; area_col; 0; 0]
      Bv.x = (lane < 16) ? 1.0f : 0.0f;
      Bv.y = (lane < 16) ? sar[jc] : 0.0f;

      v8f C;
      float inter[8];
      #pragma unroll
      for (int r = 0; r < 8; ++r) {
        float w = fminf(rx2[r], cx2) - fmaxf(rx1[r], cx1);
        float h = fminf(ry2[r], cy2) - fmaxf(ry1[r], cy1);
        w = fmaxf(w, 0.0f); h = fmaxf(h, 0.0f);
        inter[r] = w * h;
        C[r] = -inter[r];
      }
      // union tile = area_i + area_j - inter : one matrix op
      v8f U = __builtin_amdgcn_wmma_f32_16x16x4_f32(
          false, A, false, Bv, (short)0, C, false, false);

      // Gather the 8 wave-uniform ballots into lanes, single b16 store.
      unsigned sel = 0u;
      #pragma unroll
      for (int r = 0; r < 8; ++r) {
        bool adjp = (U[r] > 0.0f) && (inter[r] > NMS_THR * U[r]);  // iou>thr
        unsigned bal = __builtin_amdgcn_ballot_w32(adjp);
        if ((lane & 7) == r) sel = bal;
      }
      unsigned short hv = (lane < 8) ? (unsigned short)(sel & 0xFFFFu)
                                     : (unsigned short)(sel >> 16);
      if (lane < 16) adj16[lane][jt] = hv;   // row = lane
    }
    __syncthreads();

    // Greedy application (sequential; matches reference fori_loop semantics)
    if (tid == 0) {
      int mmax = nv - i0; if (mmax > 16) mmax = 16;
      for (int m = 0; m < mmax; ++m) {
        int i = i0 + m;
        if ((suppW[i >> 5] >> (i & 31)) & 1u) continue;
        for (int jw = i >> 4; jw < jtMax; ++jw) {
          unsigned mbits = adj16[m][jw];
          int base = jw << 4;
          if (base <= i) mbits &= ~((2u << (i - base)) - 1u);  // only j > i
          suppW[base >> 5] |= mbits << (base & 31);
        }
      }
    }
    __syncthreads();   // also protects adj16 reuse next strip
  }

  // ---- Stage 5: keep mask -------------------------------------------------
  float* keepB = keepOut + (size_t)b * NBOX;
  for (int jj = tid; jj < NBOX; jj += 256) {
    bool kept = (jj < nv) && !((suppW[jj >> 5] >> (jj & 31)) & 1u);
    keepB[jj] = kept ? 1.0f : 0.0f;
  }
}

// ---------------------------------------------------------------------------
extern "C" void kernel_launch(void* const* d_in, const int* in_sizes, int n_in,
                              void* d_out, int out_size, void* d_ws, size_t ws_size,
                              hipStream_t stream) {
  (void)in_sizes; (void)n_in; (void)out_size; (void)d_ws; (void)ws_size;
  const float* pred = (const float*)d_in[0];
  float* out      = (float*)d_out;
  float* boxesOut = out;                                  // [B,784,6]
  float* keepOut  = out + (size_t)BB * NBOX * 6;          // [B,784]

  // One block per batch; 285392 B dynamic LDS (fits 320KB/WGP on gfx1250).
  hipLaunchKernelGGL(yolo_fused_kernel, dim3(BB), dim3(256), SMEM_BYTES,
                     stream, pred, boxesOut, keepOut);
}